// CTCLayer_74603581931722
// MI455X (gfx1250) — compile-verified
//
#include <hip/hip_runtime.h>

// ---------------------------------------------------------------------------
// CTC batch loss (forward algorithm), B=512, T=1024, C=256, L=128, S=257.
// One workgroup per batch element (288 threads = 9 wave32, thread == state).
// y_pred is streamed global->LDS with CDNA5 async-to-LDS loads through a
// 4-deep ring of 8-timestep chunks, throttled with s_wait_asynccnt so ~3
// chunks of HBM latency stay hidden behind the serial alpha recurrence.
// Per-state constants (extended label channel, skip flag) live in registers.
// ---------------------------------------------------------------------------

#define AS1 __attribute__((address_space(1)))
#define AS3 __attribute__((address_space(3)))

typedef int v4i __attribute__((__vector_size__(16)));  // int4, matches builtin sig

#ifndef __has_builtin
#define __has_builtin(x) 0
#endif

#if __has_builtin(__builtin_amdgcn_global_load_async_to_lds_b128)
#define CTC_USE_ASYNC 1
#else
#define CTC_USE_ASYNC 0
#endif

#if CTC_USE_ASYNC
#if __has_builtin(__builtin_amdgcn_s_wait_asynccnt)
#define WAIT_ASYNC(n) __builtin_amdgcn_s_wait_asynccnt(n)
#else
#define WAIT_ASYNC(n) asm volatile("s_wait_asynccnt %0" ::"i"(n) : "memory")
#endif
#else
#define WAIT_ASYNC(n) ((void)0)
#endif

__device__ __forceinline__ void ctc_async_copy16(void* lds_dst, const void* gsrc) {
#if CTC_USE_ASYNC
  void* g = const_cast<void*>(gsrc);
  __builtin_amdgcn_global_load_async_to_lds_b128((AS1 v4i*)g, (AS3 v4i*)lds_dst,
                                                 /*offset=*/0, /*cpol=*/0);
#else
  *reinterpret_cast<float4*>(lds_dst) = *reinterpret_cast<const float4*>(gsrc);
#endif
}

namespace {
constexpr int kT = 1024;
constexpr int kC = 256;
constexpr int kL = 128;
constexpr int kS = 2 * kL + 1;   // 257
constexpr int kTQ = 8;           // timesteps per LDS chunk (8 KB)
constexpr int kNCH = kT / kTQ;   // 128 chunks
constexpr int kNBUF = 4;         // ring depth (32 KB of LDS staging)
constexpr float kEPS = 1e-7f;
constexpr float kNEG = -1e30f;
}  // namespace

__global__ __launch_bounds__(288, 1) void ctc_forward_kernel(
    const int* __restrict__ y_true,   // [B, L]
    const float* __restrict__ y_pred, // [B, T, C] probabilities
    float* __restrict__ out) {        // [B, 1]
  const int b = blockIdx.x;
  const int tid = threadIdx.x;
  const int blank = kC - 1;

  __shared__ float buf[kNBUF][kTQ * kC];   // 4-deep ring of y_pred chunks (32 KB)
  __shared__ float alphaBuf[2][kS + 2];    // ping-pong alpha, 2 NEG pad entries

  // ---- per-state constants, kept in registers (thread == state) ----
  int myCh = blank;   // extended label channel for my state
  int mySkip = 0;     // is (s-2 -> s) transition allowed
  if (tid < kS && (tid & 1)) {
    const int li = tid >> 1;
    myCh = y_true[(size_t)b * kL + li];
    if (li > 0) {
      const int prev = y_true[(size_t)b * kL + li - 1];
      mySkip = (myCh != blank) && (myCh != prev);
    }
  }
  if (tid < 2) {
    alphaBuf[0][tid] = kNEG;  // padding so a2/a3 reads are branch-free
    alphaBuf[1][tid] = kNEG;
  }

  const float* gbase = y_pred + (size_t)b * kT * kC;

  // ---- prologue: prefetch chunks 0..3 (8 KB each; 2 async b128 per wave/chunk)
  if (tid < 256) {
    for (int p = 0; p < kNBUF; ++p) {
      const float* src = gbase + (size_t)p * kTQ * kC + (size_t)tid * 8;
      float* dst = &buf[p][tid * 8];
      ctc_async_copy16(dst, src);
      ctc_async_copy16(dst + 4, src + 4);
    }
  }

  // chunk 0 ready: async loads complete in order; with 8 outstanding,
  // counter <= 6 retires the oldest chunk's 2 loads.
  WAIT_ASYNC(2 * (kNBUF - 1));
  __syncthreads();

  float* A = alphaBuf[0];
  float* Bp = alphaBuf[1];

  // ---- t = 0 init: only states 0 and 1 reachable ----
  float aSelf = kNEG;
  if (tid < kS) {
    const float e = __logf(buf[0][myCh] + kEPS);
    aSelf = (tid < 2) ? e : kNEG;
    A[tid + 2] = aSelf;
  }
  __syncthreads();

  // ---- serial scan over T, 4-deep async-pipelined chunks ----
  for (int c = 0; c < kNCH; ++c) {
    if (c > 0) {
      const int rem = kNCH - c;  // == chunks still in flight (incl. this one)
      if (rem >= kNBUF) {
        WAIT_ASYNC(2 * (kNBUF - 1));
      } else if (rem == 3) {
        WAIT_ASYNC(4);
      } else if (rem == 2) {
        WAIT_ASYNC(2);
      } else {
        WAIT_ASYNC(0);
      }
      __syncthreads();  // chunk c's LDS data visible to all waves
    }

    const float* cb = buf[c & (kNBUF - 1)];
    const int tl0 = (c == 0) ? 1 : 0;  // t=0 consumed by init
    for (int tl = tl0; tl < kTQ; ++tl) {
      if (tid < kS) {
        const float a1 = aSelf;                      // alpha[t-1][s]
        const float a2 = A[tid + 1];                 // alpha[t-1][s-1]
        const float a3 = mySkip ? A[tid] : kNEG;     // alpha[t-1][s-2]
        const float m = fmaxf(a1, fmaxf(a2, a3));
        const float sum = __expf(a1 - m) + __expf(a2 - m) + __expf(a3 - m);
        const float pv = cb[tl * kC + myCh];         // gathered emission
        aSelf = m + __logf(sum) + __logf(pv + kEPS);
        Bp[tid + 2] = aSelf;
      }
      __syncthreads();                // single barrier per timestep (ping-pong)
      float* tmp = A; A = Bp; Bp = tmp;
    }

    // Ring slot c%4 is free now (reads done before last barrier): refill it.
    if (c + kNBUF < kNCH && tid < 256) {
      const float* src = gbase + (size_t)(c + kNBUF) * kTQ * kC + (size_t)tid * 8;
      float* dst = &buf[c & (kNBUF - 1)][tid * 8];
      ctc_async_copy16(dst, src);
      ctc_async_copy16(dst + 4, src + 4);
    }
  }

  // ---- -logaddexp(alpha[S-1], alpha[S-2]) ----
  if (tid == 0) {
    const float x = A[kS + 1];          // state S-1
    const float y = A[kS];              // state S-2
    const float m = fmaxf(x, y);
    out[b] = -(m + __logf(__expf(x - m) + __expf(y - m)));
  }
}

extern "C" void kernel_launch(void* const* d_in, const int* in_sizes, int n_in,
                              void* d_out, int out_size, void* d_ws, size_t ws_size,
                              hipStream_t stream) {
  (void)n_in; (void)d_ws; (void)ws_size; (void)in_sizes;
  const int* y_true = (const int*)d_in[0];      // [B, L] int32
  const float* y_pred = (const float*)d_in[1];  // [B, T, C] float32
  float* out = (float*)d_out;                   // [B, 1] float32
  const int B = out_size;                       // 512
  hipLaunchKernelGGL(ctc_forward_kernel, dim3(B), dim3(288), 0, stream,
                     y_true, y_pred, out);
}